// Net_46334107189491
// MI455X (gfx1250) — compile-verified
//
#include <hip/hip_runtime.h>
#include <hip/hip_bf16.h>
#include <cstdint>
#include <cstddef>

// ---------------------------------------------------------------------------
// CDNA5 (gfx1250) fp32 WMMA helpers. wave32 only.
// A 16x4 fp32 layout: lanes 0-15 -> M=lane, K={k0,k0+1}; lanes 16-31 -> K={k0+2,k0+3}
// C/D 16x16 fp32 layout: lanes 0-15 -> N=lane, M=v; lanes 16-31 -> M=v+8
// ---------------------------------------------------------------------------
typedef float v2f __attribute__((ext_vector_type(2)));
typedef float v8f __attribute__((ext_vector_type(8)));

#if defined(__HIP_DEVICE_COMPILE__)
#if !__has_builtin(__builtin_amdgcn_wmma_f32_16x16x4_f32)
#error "gfx1250 fp32 WMMA builtin (__builtin_amdgcn_wmma_f32_16x16x4_f32) not available"
#endif
#endif

#define HIDF 160
#define NPFC 8192
#define NVTX 2048
#define KNN1 32
#define KNN2 16
#define CFPAD 172   // 171 logical features padded to multiple of 4
#define BIGD 1e30f

// Fast reciprocal: single v_rcp_f32 (~1 ulp) instead of the IEEE div expansion.
__device__ __forceinline__ float dev_rcp(float x) {
#if defined(__HIP_DEVICE_COMPILE__)
  return __builtin_amdgcn_rcpf(x);
#else
  return 1.0f / x;
#endif
}
__device__ __forceinline__ float dev_sigmoid(float x) { return dev_rcp(1.0f + __expf(-x)); }
__device__ __forceinline__ float dev_silu(float x)    { return x * dev_sigmoid(x); }

// D = A(16 x kdim) * B(kdim x 16) + cinit, via chained v_wmma_f32_16x16x4_f32.
// fa(m,k) supplies A, fb(k,n) supplies B for this lane's (m=n=lane&15) slice.
template <typename FA, typename FB>
__device__ __forceinline__ v8f wmma_mac16(int kdim, FA fa, FB fb, float cinit) {
  const int lane = threadIdx.x & 31;
  const int hi   = lane >> 4;
  const int mn   = lane & 15;
  v8f acc;
#pragma unroll
  for (int v = 0; v < 8; ++v) acc[v] = cinit;
  for (int k0 = 0; k0 < kdim; k0 += 4) {
    const int bk = k0 + 2 * hi;
    v2f a, b;
    a.x = fa(mn, bk);
    a.y = fa(mn, bk + 1);
    b.x = fb(bk, mn);
    b.y = fb(bk + 1, mn);
    acc = __builtin_amdgcn_wmma_f32_16x16x4_f32(false, a, false, b, (short)0, acc,
                                                false, false);
  }
  return acc;
}

// ---------------------------------------------------------------------------
// Particle encoder: enc = charged_mlp(x12)*cm + neutral_mlp(x11)*(1-cm)
// One wave per 16 particles.  12->80->160 (charged), 11->80->160 (neutral).
// ---------------------------------------------------------------------------
__global__ void pfc_encode_kernel(const float* __restrict__ x,
                                  const float* __restrict__ wc0, const float* __restrict__ bc0,
                                  const float* __restrict__ wc1, const float* __restrict__ bc1,
                                  const float* __restrict__ wn0, const float* __restrict__ bn0,
                                  const float* __restrict__ wn1, const float* __restrict__ bn1,
                                  float* __restrict__ enc) {
  __shared__ float ls_x[16][12];
  __shared__ float ls_h[16][80];
  __shared__ float ls_oc[16][HIDF];
  const int r0   = blockIdx.x * 16;
  const int tid  = threadIdx.x;
  const int lane = tid & 31;
  const int col  = lane & 15;
  const int hi   = lane >> 4;

  for (int e = tid; e < 16 * 12; e += 32) {
    ls_x[e / 12][e % 12] = x[(size_t)(r0 + e / 12) * 12 + (e % 12)];
  }
  __syncthreads();

  // charged layer 1: K=12, N=80
  for (int nt = 0; nt < 5; ++nt) {
    const int c = nt * 16 + col;
    v8f acc = wmma_mac16(12,
        [&](int m, int k) { return ls_x[m][k]; },
        [&](int k, int n) { return wc0[k * 80 + c]; },
        bc0[c]);
#pragma unroll
    for (int v = 0; v < 8; ++v) ls_h[v + 8 * hi][c] = dev_silu(acc[v]);
  }
  __syncthreads();
  // charged layer 2: K=80, N=160 -> ls_oc
  for (int nt = 0; nt < 10; ++nt) {
    const int c = nt * 16 + col;
    v8f acc = wmma_mac16(80,
        [&](int m, int k) { return ls_h[m][k]; },
        [&](int k, int n) { return wc1[k * HIDF + c]; },
        bc1[c]);
#pragma unroll
    for (int v = 0; v < 8; ++v) ls_oc[v + 8 * hi][c] = acc[v];
  }
  __syncthreads();
  // neutral layer 1: K=11 (padded to 12), N=80
  for (int nt = 0; nt < 5; ++nt) {
    const int c = nt * 16 + col;
    v8f acc = wmma_mac16(12,
        [&](int m, int k) { return (k < 11) ? ls_x[m][k] : 0.0f; },
        [&](int k, int n) { return (k < 11) ? wn0[k * 80 + c] : 0.0f; },
        bn0[c]);
#pragma unroll
    for (int v = 0; v < 8; ++v) ls_h[v + 8 * hi][c] = dev_silu(acc[v]);
  }
  __syncthreads();
  // neutral layer 2 + blend by charge (x col 10)
  for (int nt = 0; nt < 10; ++nt) {
    const int c = nt * 16 + col;
    v8f acc = wmma_mac16(80,
        [&](int m, int k) { return ls_h[m][k]; },
        [&](int k, int n) { return wn1[k * HIDF + c]; },
        bn1[c]);
#pragma unroll
    for (int v = 0; v < 8; ++v) {
      const int m = v + 8 * hi;
      const bool charged = (ls_x[m][10] != 0.0f);
      enc[(size_t)(r0 + m) * HIDF + c] = charged ? ls_oc[m][c] : acc[v];
    }
  }
}

// ---------------------------------------------------------------------------
// Row squared norms.
// ---------------------------------------------------------------------------
__global__ void sqnorm_kernel(const float* __restrict__ X, int fs, int fdim,
                              float* __restrict__ sqn, int n) {
  const int i = blockIdx.x * blockDim.x + threadIdx.x;
  if (i >= n) return;
  float s = 0.0f;
  for (int c = 0; c < fdim; ++c) {
    const float v = X[(size_t)i * fs + c];
    s += v * v;
  }
  sqn[i] = s;
}

// ---------------------------------------------------------------------------
// Fused kNN: one wave owns 16 query rows; Gram tiles via WMMA; streaming
// replace-worst top-k in LDS.  chargecol!=nullptr restricts sources to charged.
// ---------------------------------------------------------------------------
__global__ void knn_kernel(const float* __restrict__ X, int fs, int fdim,
                           const float* __restrict__ sqn,
                           const int* __restrict__ batch,
                           const float* __restrict__ chargecol,
                           int n, int k, int* __restrict__ idxout) {
  __shared__ float ls_A[16][CFPAD + 4];
  __shared__ float ls_nd[16];
  __shared__ int   ls_bd[16];
  __shared__ float ls_d[16][16];
  __shared__ float ls_td[16][KNN1];
  __shared__ int   ls_ti[16][KNN1];

  const int r0   = blockIdx.x * 16;
  const int tid  = threadIdx.x;
  const int lane = tid & 31;
  const int col  = lane & 15;
  const int hi   = lane >> 4;

  for (int e = tid; e < 16 * fdim; e += 32) {
    ls_A[e / fdim][e % fdim] = X[(size_t)(r0 + e / fdim) * fs + (e % fdim)];
  }
  if (tid < 16) {
    ls_nd[tid] = sqn[r0 + tid];
    ls_bd[tid] = batch[r0 + tid];
    for (int s = 0; s < k; ++s) { ls_td[tid][s] = BIGD; ls_ti[tid][s] = 0; }
  }
  float wval = BIGD;  // current worst in this row's list (thread tid<16 owns row tid)
  int   wslot = 0;
  __syncthreads();

  for (int j0 = 0; j0 < n; j0 += 16) {
    v8f acc = wmma_mac16(fdim,
        [&](int m, int kk) { return ls_A[m][kk]; },
        [&](int kk, int nn) { return X[(size_t)(j0 + col) * fs + kk]; },
        0.0f);
#pragma unroll
    for (int v = 0; v < 8; ++v) {
      const int m = v + 8 * hi;
      const int j = j0 + col;
      float d = ls_nd[m] + sqn[j] - 2.0f * acc[v];
      bool bad = (batch[j] != ls_bd[m]);
      if (chargecol) bad = bad || (chargecol[(size_t)j * 12 + 10] == 0.0f);
      ls_d[m][col] = bad ? BIGD : d;
    }
    __syncthreads();
    if (tid < 16) {
      for (int nn = 0; nn < 16; ++nn) {
        const float d = ls_d[tid][nn];
        if (d < wval) {
          ls_td[tid][wslot] = d;
          ls_ti[tid][wslot] = j0 + nn;
          float mx = -BIGD; int ms = 0;
          for (int s = 0; s < k; ++s) {
            const float t = ls_td[tid][s];
            if (t > mx) { mx = t; ms = s; }
          }
          wval = mx; wslot = ms;
        }
      }
    }
    __syncthreads();
  }
  if (tid < 16) {
    for (int s = 0; s < k; ++s) idxout[(size_t)(r0 + tid) * k + s] = ls_ti[tid][s];
  }
}

// ---------------------------------------------------------------------------
// cf = [x_pfc[:, :11], feats1] padded to stride 172 with a zero column.
// ---------------------------------------------------------------------------
__global__ void build_cf_kernel(const float* __restrict__ x,
                                const float* __restrict__ feats1,
                                float* __restrict__ cf) {
  const int t = blockIdx.x * blockDim.x + threadIdx.x;
  if (t >= NPFC * CFPAD) return;
  const int i = t / CFPAD, c = t % CFPAD;
  float v;
  if (c < 11)        v = x[(size_t)i * 12 + c];
  else if (c < 171)  v = feats1[(size_t)i * HIDF + (c - 11)];
  else               v = 0.0f;
  cf[t] = v;
}

// ---------------------------------------------------------------------------
// Fused EdgeConv: one workgroup per destination node; blockDim = 32*(k/16).
// layer1 (2F -> 160, SiLU) operands gathered as [xi, xj-xi]; hidden in LDS;
// layer2 (160 -> 160); mean over k via shfl_xor + LDS float atomics.
// khalf = padded half-width (A pad cols are zero; guarded weight rows).
// ---------------------------------------------------------------------------
__global__ void edge_conv_kernel(const float* __restrict__ X, int fs, int fvalid, int khalf,
                                 const int* __restrict__ idx, int k,
                                 const float* __restrict__ w0, const float* __restrict__ b0,
                                 const float* __restrict__ w1, const float* __restrict__ b1,
                                 float* __restrict__ out, int outstride, float scale) {
  __shared__ float ls_h[2][16][HIDF];
  __shared__ float ls_out[HIDF];
  const int dst  = blockIdx.x;
  const int tid  = threadIdx.x;
  const int wave = tid >> 5;
  const int lane = tid & 31;
  const int col  = lane & 15;
  const int hi   = lane >> 4;
  const int m    = lane & 15;

  __builtin_prefetch(w0, 0, 1);
  __builtin_prefetch(w1, 0, 1);

  for (int c = tid; c < HIDF; c += blockDim.x) ls_out[c] = 0.0f;

  const int j = idx[(size_t)dst * k + wave * 16 + m];
  const float* pi = X + (size_t)dst * fs;
  const float* pj = X + (size_t)j * fs;
  __syncthreads();

  const int kdim = 2 * khalf;
  for (int nt = 0; nt < 10; ++nt) {
    const int c = nt * 16 + col;
    v8f acc = wmma_mac16(kdim,
        [&](int mm, int kk) {
          return (kk < khalf) ? pi[kk] : (pj[kk - khalf] - pi[kk - khalf]);
        },
        [&](int kk, int nn) {
          const int  kh = (kk < khalf) ? kk : (kk - khalf);
          const bool ok = (kh < fvalid);
          const int  wr = (kk < khalf) ? kh : (fvalid + kh);
          return ok ? w0[(size_t)wr * HIDF + c] : 0.0f;
        },
        b0[c]);
#pragma unroll
    for (int v = 0; v < 8; ++v) ls_h[wave][v + 8 * hi][c] = dev_silu(acc[v]);
  }
  __syncthreads();

  for (int nt = 0; nt < 10; ++nt) {
    const int c = nt * 16 + col;
    v8f acc = wmma_mac16(HIDF,
        [&](int mm, int kk) { return ls_h[wave][mm][kk]; },
        [&](int kk, int nn) { return w1[(size_t)kk * HIDF + c]; },
        b1[c]);
    float part = 0.0f;
#pragma unroll
    for (int v = 0; v < 8; ++v) part += acc[v];
    part += __shfl_xor(part, 16, 32);      // combine row halves (wave32)
    if (lane < 16) atomicAdd(&ls_out[c], part);  // ds_add_f32
  }
  __syncthreads();
  for (int c = tid; c < HIDF; c += blockDim.x) {
    out[(size_t)dst * outstride + c] = ls_out[c] * scale;
  }
}

// ---------------------------------------------------------------------------
// Output head: feats2 -> 40 (SiLU) -> 4 (SiLU) -> 1 (sigmoid); also emits
// batch_pfc as float.
// ---------------------------------------------------------------------------
__global__ void out_head_kernel(const float* __restrict__ feats2,
                                const int* __restrict__ batch,
                                const float* __restrict__ w0, const float* __restrict__ b0,
                                const float* __restrict__ w1, const float* __restrict__ b1,
                                const float* __restrict__ w2, const float* __restrict__ b2,
                                float* __restrict__ out0, float* __restrict__ outb) {
  const int i = blockIdx.x * blockDim.x + threadIdx.x;
  if (i >= NPFC) return;
  float h1[40];
  for (int o = 0; o < 40; ++o) {
    float s = b0[o];
    for (int c = 0; c < HIDF; ++c) s += feats2[(size_t)i * HIDF + c] * w0[c * 40 + o];
    h1[o] = dev_silu(s);
  }
  float h2[4];
  for (int o = 0; o < 4; ++o) {
    float s = b1[o];
    for (int c = 0; c < 40; ++c) s += h1[c] * w1[c * 4 + o];
    h2[o] = dev_silu(s);
  }
  float s = b2[0];
  for (int c = 0; c < 4; ++c) s += h2[c] * w2[c];
  out0[i] = dev_sigmoid(s);
  outb[i] = (float)batch[i];
}

// ---------------------------------------------------------------------------
// Vertex encoder: 5 -> 40 -> 80 -> 160 (SiLU between layers). Tiny; VALU.
// ---------------------------------------------------------------------------
__global__ void vtx_kernel(const float* __restrict__ xv,
                           const float* __restrict__ w0, const float* __restrict__ b0,
                           const float* __restrict__ w1, const float* __restrict__ b1,
                           const float* __restrict__ w2, const float* __restrict__ b2,
                           float* __restrict__ out) {
  const int i = blockIdx.x * blockDim.x + threadIdx.x;
  if (i >= NVTX) return;
  float h1[40];
  for (int o = 0; o < 40; ++o) {
    float s = b0[o];
    for (int c = 0; c < 5; ++c) s += xv[(size_t)i * 5 + c] * w0[c * 40 + o];
    h1[o] = dev_silu(s);
  }
  float h2[80];
  for (int o = 0; o < 80; ++o) {
    float s = b1[o];
    for (int c = 0; c < 40; ++c) s += h1[c] * w1[c * 80 + o];
    h2[o] = dev_silu(s);
  }
  for (int o = 0; o < HIDF; ++o) {
    float s = b2[o];
    for (int c = 0; c < 80; ++c) s += h2[c] * w2[c * HIDF + o];
    out[(size_t)i * HIDF + o] = s;
  }
}

// ---------------------------------------------------------------------------
// Host: locate inputs by unique flat element counts (robust to pytree order),
// carve workspace, launch the pipeline on `stream`.
// ---------------------------------------------------------------------------
static int find_size(const int* s, int n, int sz) {
  for (int i = 0; i < n; ++i) if (s[i] == sz) return i;
  return -1;
}

extern "C" void kernel_launch(void* const* d_in, const int* in_sizes, int n_in,
                              void* d_out, int out_size, void* d_ws, size_t ws_size,
                              hipStream_t stream) {
  // --- locate inputs/params by unique flat sizes ---
  const float* x_pfc   = (const float*)d_in[find_size(in_sizes, n_in, NPFC * 12)];   // 98304
  const float* x_vtx   = (const float*)d_in[find_size(in_sizes, n_in, NVTX * 5)];    // 10240
  const int*   b_pfc   = (const int*)  d_in[find_size(in_sizes, n_in, NPFC)];        // 8192
  const int iv = find_size(in_sizes, n_in, 200);     // vtx w0 [5,40]
  const int in_ = find_size(in_sizes, n_in, 880);    // neutral w0 [11,80]
  const int ic = find_size(in_sizes, n_in, 960);     // charged w0 [12,80]
  const int i1 = find_size(in_sizes, n_in, 51200);   // conv1 w0 [320,160]
  const int i2 = find_size(in_sizes, n_in, 54720);   // conv2 w0 [342,160]
  const int io = find_size(in_sizes, n_in, 6400);    // out w0 [160,40]
  #define P(base, off) ((const float*)d_in[(base) + (off)])

  // --- workspace carve ---
  char* ws = (char*)d_ws;
  size_t off = 0;
  auto carve = [&](size_t bytes) { char* p = ws + off; off += (bytes + 255) & ~size_t(255); return p; };
  float* enc    = (float*)carve((size_t)NPFC * HIDF * 4);
  float* cf     = (float*)carve((size_t)NPFC * CFPAD * 4);
  float* feats2 = (float*)carve((size_t)NPFC * HIDF * 4);
  float* sqn1   = (float*)carve((size_t)NPFC * 4);
  float* sqn2   = (float*)carve((size_t)NPFC * 4);
  int*   idx1   = (int*)  carve((size_t)NPFC * KNN1 * 4);
  int*   idx2   = (int*)  carve((size_t)NPFC * KNN2 * 4);
  (void)ws_size;

  // --- output layout: out(8192) | batch(8192) | feats1(8192*160) | vtx_enc(2048*160) ---
  float* out0    = (float*)d_out;
  float* outb    = out0 + NPFC;
  float* feats1  = outb + NPFC;
  float* vtx_enc = feats1 + (size_t)NPFC * HIDF;
  (void)out_size;

  // 1) vertex encoder
  vtx_kernel<<<NVTX / 256, 256, 0, stream>>>(x_vtx, P(iv,0), P(iv,1), P(iv,2), P(iv,3),
                                             P(iv,4), P(iv,5), vtx_enc);
  // 2) particle encoder (WMMA)
  pfc_encode_kernel<<<NPFC / 16, 32, 0, stream>>>(x_pfc,
      P(ic,0), P(ic,1), P(ic,2), P(ic,3),
      P(in_,0), P(in_,1), P(in_,2), P(in_,3), enc);
  // 3) kNN-1 over enc (K=32)
  sqnorm_kernel<<<NPFC / 256, 256, 0, stream>>>(enc, HIDF, HIDF, sqn1, NPFC);
  knn_kernel<<<NPFC / 16, 32, 0, stream>>>(enc, HIDF, HIDF, sqn1, b_pfc, nullptr,
                                           NPFC, KNN1, idx1);
  // 4) EdgeConv-1 (WMMA) -> feats1 (direct to d_out)
  edge_conv_kernel<<<NPFC, 64, 0, stream>>>(enc, HIDF, HIDF, HIDF, idx1, KNN1,
      P(i1,0), P(i1,1), P(i1,2), P(i1,3), feats1, HIDF, 1.0f / KNN1);
  // 5) cf = [x[:, :11], feats1] padded
  build_cf_kernel<<<(NPFC * CFPAD + 255) / 256, 256, 0, stream>>>(x_pfc, feats1, cf);
  // 6) kNN-2 over cf (K=16, sources restricted to charged)
  sqnorm_kernel<<<NPFC / 256, 256, 0, stream>>>(cf, CFPAD, CFPAD, sqn2, NPFC);
  knn_kernel<<<NPFC / 16, 32, 0, stream>>>(cf, CFPAD, CFPAD, sqn2, b_pfc, x_pfc,
                                           NPFC, KNN2, idx2);
  // 7) EdgeConv-2 (WMMA) -> feats2
  edge_conv_kernel<<<NPFC, 32, 0, stream>>>(cf, CFPAD, 171, CFPAD, idx2, KNN2,
      P(i2,0), P(i2,1), P(i2,2), P(i2,3), feats2, HIDF, 1.0f / KNN2);
  // 8) output head + batch passthrough
  out_head_kernel<<<NPFC / 256, 256, 0, stream>>>(feats2, b_pfc,
      P(io,0), P(io,1), P(io,2), P(io,3), P(io,4), P(io,5), out0, outb);
  #undef P
}